// PdfSampler_63170378989664
// MI455X (gfx1250) — compile-verified
//
#include <hip/hip_runtime.h>
#include <cstdint>

typedef float v2f __attribute__((ext_vector_type(2)));
typedef float v4f __attribute__((ext_vector_type(4)));
typedef float v8f __attribute__((ext_vector_type(8)));

#define WAVES            4
#define RAYS_PER_WAVE    16
#define RAYS_PER_BLOCK   (WAVES * RAYS_PER_WAVE)   // 64 rays / block
#define NB               128                       // bins & samples
#define ROWSTRIDE        132                       // 128 cdf slots + pad (bank stagger)
#define TINYF            1e-6f

// One wave32 = 16 rays. CDF prefix sum done on the matrix core:
//   per 16-wide chunk: P_chunk = W_chunk(16x16) x U16 (upper-tri ones),
//   as 4 x V_WMMA_F32_16X16X4_F32 accumulations.
__global__ __launch_bounds__(WAVES * 32) void pdf_sampler_kernel(
    const float* __restrict__ rays_o,
    const float* __restrict__ rays_d,
    const float* __restrict__ weights,
    float* __restrict__ out,
    int Btot)
{
    __shared__ float sW[RAYS_PER_BLOCK * ROWSTRIDE];   // weights, overlaid by cdf
    __shared__ float sOff[RAYS_PER_BLOCK * 9];         // 8 chunk offsets + 1/total

    const int tid      = threadIdx.x;
    const int lane     = tid & 31;
    const int wave     = tid >> 5;
    const int blockRay = blockIdx.x * RAYS_PER_BLOCK;

    // ---- stage weights (+TINY) into LDS, float4 coalesced, NT (read-once) ----
    {
        const v4f* g = (const v4f*)(weights + (size_t)blockRay * NB);
        #pragma unroll
        for (int j = 0; j < 16; ++j) {
            int f   = tid + (WAVES * 32) * j;     // 0..2047 float4s
            int ray = f >> 5;                     // (f*4)/128
            int col = (f & 31) << 2;
            v4f v = __builtin_nontemporal_load(&g[f]);
            v4f w = v + TINYF;
            *(v4f*)&sW[ray * ROWSTRIDE + col] = w;
        }
    }
    __syncthreads();

    float* myW   = &sW[wave * RAYS_PER_WAVE * ROWSTRIDE];
    float* myOff = &sOff[wave * RAYS_PER_WAVE * 9];

    const int col  = lane & 15;   // N for B/D frags; row M for A frags
    const int half = lane >> 4;

    // ---- B fragments of U16 (inclusive upper-triangular ones), K-steps 0..3 --
    v2f bfrag[4];
    #pragma unroll
    for (int s = 0; s < 4; ++s) {
        int k0 = 4 * s + 2 * half;
        bfrag[s].x = (k0     <= col) ? 1.0f : 0.0f;
        bfrag[s].y = (k0 + 1 <= col) ? 1.0f : 0.0f;
    }

    // ---- within-chunk prefix sums via WMMA (reverse chunks: safe in-place) ---
    for (int kt = 7; kt >= 0; --kt) {
        v8f acc = {};
        #pragma unroll
        for (int s = 0; s < 4; ++s) {
            // A frag: row = lane&15, cols kt*16 + 4s + 2*half .. +1  (ds_load_b64)
            const float* ap = &myW[col * ROWSTRIDE + kt * 16 + 4 * s + 2 * half];
            v2f a = *(const v2f*)ap;
            acc = __builtin_amdgcn_wmma_f32_16x16x4_f32(
                false, a, false, bfrag[s], (short)0, acc, false, false);
        }
        // D tile element (M = v + 8*half, N = col) -> cdf slot 1 + kt*16 + col
        #pragma unroll
        for (int v = 0; v < 8; ++v) {
            myW[(v + 8 * half) * ROWSTRIDE + 1 + kt * 16 + col] = acc[v];
        }
    }
    __syncthreads();

    // ---- cross-chunk exclusive scan + 1/total, lane r handles ray r ----------
    if (lane < 16) {
        const int r = lane;
        float run = 0.0f;
        #pragma unroll
        for (int kt = 0; kt < 8; ++kt) {
            myOff[r * 9 + kt] = run;
            run += myW[r * ROWSTRIDE + 16 * (kt + 1)];   // chunk total
        }
        myOff[r * 9 + 8] = 1.0f / run;
        myW[r * ROWSTRIDE + 0] = 0.0f;                    // cdf[0] = 0
    }
    __syncthreads();

    // ---- normalize: cdf = (within + offset) / total --------------------------
    #pragma unroll 4
    for (int i = 0; i < 64; ++i) {
        int idx = i * 32 + lane;            // 0..2047
        int r   = idx >> 7;
        int c   = (idx & 127) + 1;          // 1..128
        float off = myOff[r * 9 + ((c - 1) >> 4)];
        float inv = myOff[r * 9 + 8];
        myW[r * ROWSTRIDE + c] = (myW[r * ROWSTRIDE + c] + off) * inv;
    }
    __syncthreads();

    // ---- invert CDF at u = i/127, interpolate analytic bins, write outputs ---
    const float DSTEP = 49.0f / 127.0f;     // bin spacing
    const float UINV  = 1.0f / 127.0f;
    const size_t zBase = (size_t)Btot * 384;
    const size_t sBase = (size_t)Btot * 512;

    for (int r = 0; r < RAYS_PER_WAVE; ++r) {
        const size_t rayG = (size_t)(blockRay + wave * RAYS_PER_WAVE + r);
        const float* cdf  = &myW[r * ROWSTRIDE];
        const float ox = rays_o[rayG * 3 + 0], oy = rays_o[rayG * 3 + 1], oz = rays_o[rayG * 3 + 2];
        const float dx = rays_d[rayG * 3 + 0], dy = rays_d[rayG * 3 + 1], dz = rays_d[rayG * 3 + 2];

        float sv[4];
        #pragma unroll
        for (int q = 0; q < 4; ++q) {
            const int   i = lane * 4 + q;           // lane owns 4 contiguous samples
            const float u = (float)i * UINV;
            // searchsorted(cdf[0:128], u, side='right') -> first idx with cdf>u
            int lo = 0, hi = 128;
            #pragma unroll
            for (int it = 0; it < 7; ++it) {
                int  mid = (lo + hi) >> 1;
                bool le  = cdf[mid] <= u;
                lo = le ? mid + 1 : lo;
                hi = le ? hi : mid;
            }
            const int above = lo;                          // [0,128]
            const int below = above > 0 ? above - 1 : 0;
            const float clo = cdf[below];
            const float chi = cdf[above];
            const float blo = (below == 0) ? 1.0f
                            : (below == 128) ? 50.0f
                            : 1.0f + ((float)below - 0.5f) * DSTEP;
            const float bhi = (above == 0) ? 1.0f
                            : (above == 128) ? 50.0f
                            : 1.0f + ((float)above - 0.5f) * DSTEP;
            float denom = chi - clo;
            if (denom < TINYF) denom = 1.0f;
            const float t = (u - clo) / denom;
            sv[q] = blo + t * (bhi - blo + TINYF);
        }

        // z and s (identical; u ascending + monotone cdf => already sorted)
        // NT stores: write-once streaming data, keep it out of L2 residency.
        const v4f zq = { sv[0], sv[1], sv[2], sv[3] };
        const size_t zoff = rayG * 128 + (size_t)lane * 4;
        __builtin_nontemporal_store(zq, (v4f*)(out + zBase + zoff));
        __builtin_nontemporal_store(zq, (v4f*)(out + sBase + zoff));

        // pts = o + d * s : 12 consecutive floats per lane -> 3 float4 stores
        float p[12];
        #pragma unroll
        for (int q = 0; q < 4; ++q) {
            p[q * 3 + 0] = fmaf(dx, sv[q], ox);
            p[q * 3 + 1] = fmaf(dy, sv[q], oy);
            p[q * 3 + 2] = fmaf(dz, sv[q], oz);
        }
        float* pb = out + rayG * 384 + (size_t)lane * 12;
        const v4f p0 = { p[0], p[1], p[2],  p[3]  };
        const v4f p1 = { p[4], p[5], p[6],  p[7]  };
        const v4f p2 = { p[8], p[9], p[10], p[11] };
        __builtin_nontemporal_store(p0, (v4f*)(pb + 0));
        __builtin_nontemporal_store(p1, (v4f*)(pb + 4));
        __builtin_nontemporal_store(p2, (v4f*)(pb + 8));
    }
}

extern "C" void kernel_launch(void* const* d_in, const int* in_sizes, int n_in,
                              void* d_out, int out_size, void* d_ws, size_t ws_size,
                              hipStream_t stream) {
    (void)n_in; (void)out_size; (void)d_ws; (void)ws_size;
    const float* rays_o  = (const float*)d_in[0];
    const float* rays_d  = (const float*)d_in[1];
    const float* weights = (const float*)d_in[2];
    float* out = (float*)d_out;

    const int Btot   = in_sizes[2] / NB;          // 262144
    const int blocks = Btot / RAYS_PER_BLOCK;     // 4096
    hipLaunchKernelGGL(pdf_sampler_kernel, dim3(blocks), dim3(WAVES * 32), 0, stream,
                       rays_o, rays_d, weights, out, Btot);
}